// HMM_ForwardBackward_6760278524201
// MI455X (gfx1250) — compile-verified
//
#include <hip/hip_runtime.h>
#include <hip/hip_bf16.h>
#include <math.h>

typedef float v2f __attribute__((ext_vector_type(2)));
typedef float v8f __attribute__((ext_vector_type(8)));

#define N_STATES 16
#define LATENT   64
#define IN_DIM   256
#define BATCH    16
#define SEQ      2048
#define ROWS     (BATCH * SEQ)          // 32768
#define TILES    (ROWS / 16)            // 2048
#define WAVES_PER_BLOCK 8
#define EM_BLOCKS (TILES / WAVES_PER_BLOCK)  // 256

// Chunked scan config
#define CHUNK   64
#define NCHUNK  (SEQ / CHUNK)           // 32

// LDS partition (floats) for emissions kernel
#define S_WZ 0                 // 64*256 = 16384
#define S_WS 16384             // 16*256 = 4096
#define S_A  20480             // 64*16  = 1024
#define S_B2 21504             // 64*16  = 1024
#define S_BD 22528             // 256
#define S_C  22784             // 16
#define SMEM_FLOATS 22800

// Workspace layout (floats)
#define WS_EM   0
#define WS_AM   (ROWS * N_STATES)              // 524288
#define WS_B2   (WS_AM + LATENT * N_STATES)    // +1024
#define WS_CC   (WS_B2 + LATENT * N_STATES)    // +1024
#define WS_P    (WS_CC + N_STATES)             // +16 -> 526352
#define WS_BND  (WS_P + BATCH * NCHUNK * 256)  // +131072 -> 657424

__device__ __forceinline__ v8f wmma_f32(v2f a, v2f b, v8f c) {
    // D(16x16,f32) += A(16x4,f32) * B(4x16,f32)
    return __builtin_amdgcn_wmma_f32_16x16x4_f32(false, a, false, b,
                                                 (short)0, c, false, false);
}

// ---------------------------------------------------------------------------
// Setup: precompute small state matrices.
//   inv[k,l]  = 1/(exp(lv[k,l]) + 1e-8)
//   A [l,k]   = -0.5*inv[k,l]
//   B2[l,k]   = mu[k,l]*inv[k,l] - sum_i Wz[l,i]*Ws[k,i]
//   Cc[k]     = -0.5*(||Ws_k||^2 + sum_l mu^2*inv + sum_l lv + 64*ln(2pi))
//               - sum_i b_dec[i]*Ws[k,i]
// ---------------------------------------------------------------------------
__global__ void hmm_setup(const float* __restrict__ mu, const float* __restrict__ lv,
                          const float* __restrict__ Wz, const float* __restrict__ Ws,
                          const float* __restrict__ bdec,
                          float* __restrict__ Amat, float* __restrict__ B2,
                          float* __restrict__ Cc) {
    const int t = threadIdx.x;
    const int l = t >> 4;      // 0..63
    const int k = t & 15;      // 0..15
    const float lvkl = lv[k * LATENT + l];
    const float invv = 1.0f / (__expf(lvkl) + 1e-8f);
    Amat[l * N_STATES + k] = -0.5f * invv;
    float g = 0.0f;
    for (int i = 0; i < IN_DIM; ++i)
        g = fmaf(Wz[l * IN_DIM + i], Ws[k * IN_DIM + i], g);
    B2[l * N_STATES + k] = mu[k * LATENT + l] * invv - g;

    if (l == 0) {
        float s = 0.0f;  // ||Ws_k||^2 + 2*b_dec.Ws_k
        for (int i = 0; i < IN_DIM; ++i) {
            const float w = Ws[k * IN_DIM + i];
            s = fmaf(w, w, s);
            s = fmaf(2.0f * bdec[i], w, s);
        }
        float s2 = 0.0f; // sum_l mu^2*inv + lv
        for (int ll = 0; ll < LATENT; ++ll) {
            const float lvv = lv[k * LATENT + ll];
            const float iv = 1.0f / (__expf(lvv) + 1e-8f);
            const float mv = mu[k * LATENT + ll];
            s2 += mv * mv * iv + lvv;
        }
        Cc[k] = -0.5f * (s + s2 + 64.0f * 1.8378770664093453f /* ln(2*pi) */);
    }
}

// ---------------------------------------------------------------------------
// Emissions: em[row,k] = (y@Ws^T + z@B2 + z^2@A)[row,k] - 0.5*||d_row||^2 + Cc[k]
// where d = y - z@W_z - b_dec.  One wave per 16-row tile; all GEMMs via
// V_WMMA_F32_16X16X4_F32, dual accumulator chains to break C->D RAW serialization.
// ---------------------------------------------------------------------------
__global__ void hmm_emissions(const float* __restrict__ y, const float* __restrict__ z,
                              const float* __restrict__ Wz, const float* __restrict__ Ws,
                              const float* __restrict__ bdec,
                              const float* __restrict__ Amat, const float* __restrict__ B2,
                              const float* __restrict__ Cc,
                              float* __restrict__ em) {
    extern __shared__ float smem[];
    float* sWz = smem + S_WZ;
    float* sWs = smem + S_WS;
    float* sA  = smem + S_A;
    float* sB2 = smem + S_B2;
    float* sBd = smem + S_BD;
    float* sC  = smem + S_C;

    for (int i = threadIdx.x; i < LATENT * IN_DIM; i += blockDim.x)   sWz[i] = Wz[i];
    for (int i = threadIdx.x; i < N_STATES * IN_DIM; i += blockDim.x) sWs[i] = Ws[i];
    for (int i = threadIdx.x; i < LATENT * N_STATES; i += blockDim.x) {
        sA[i] = Amat[i]; sB2[i] = B2[i];
    }
    for (int i = threadIdx.x; i < IN_DIM; i += blockDim.x) sBd[i] = bdec[i];
    if (threadIdx.x < N_STATES) sC[threadIdx.x] = Cc[threadIdx.x];
    __syncthreads();

    const int lane = threadIdx.x & 31;
    const int wave = threadIdx.x >> 5;
    const int tile = blockIdx.x * WAVES_PER_BLOCK + wave;
    const int row0 = tile * 16;
    const int m  = lane & 15;   // A: row index; B/C: column (state) index
    const int hi = lane >> 4;   // K-half selector for A/B frags

    // z A-fragments: lane m holds z[row0+m, 4c + 2*hi (+1)]
    const float* zrow = z + (size_t)(row0 + m) * LATENT + hi * 2;
    v2f zf[16];
#pragma unroll
    for (int c = 0; c < 16; ++c) zf[c] = *(const v2f*)(zrow + c * 4);

    v8f acc0 = {};   // independent accumulator chains
    v8f acc1 = {};
    // prior-term GEMMs over K=64: acc += z@B2 + z^2@A  (split across chains)
#pragma unroll
    for (int c = 0; c < 16; ++c) {
        const int l = c * 4 + hi * 2;
        v2f b2f; b2f.x = sB2[l * N_STATES + m]; b2f.y = sB2[(l + 1) * N_STATES + m];
        v2f baf; baf.x = sA [l * N_STATES + m]; baf.y = sA [(l + 1) * N_STATES + m];
        acc0 = wmma_f32(zf[c], b2f, acc0);
        acc1 = wmma_f32(zf[c] * zf[c], baf, acc1);
    }

    // acc += y @ Ws^T over K=256  (B[i,k] = Ws[k,i]: contiguous pair per lane);
    // alternate chains every chunk
    const float* yrow  = y + (size_t)(row0 + m) * IN_DIM + hi * 2;
    const float* wsrow = sWs + m * IN_DIM + hi * 2;
#pragma unroll 4
    for (int c = 0; c < 64; c += 2) {
        v2f af0 = *(const v2f*)(yrow + c * 4);
        v2f bf0 = *(const v2f*)(wsrow + c * 4);
        v2f af1 = *(const v2f*)(yrow + c * 4 + 4);
        v2f bf1 = *(const v2f*)(wsrow + c * 4 + 4);
        acc0 = wmma_f32(af0, bf0, acc0);
        acc1 = wmma_f32(af1, bf1, acc1);
    }

    // ||d||^2: base tiles in C layout (dual chains), y loaded in matching C layout
    float sq[8] = {0.f, 0.f, 0.f, 0.f, 0.f, 0.f, 0.f, 0.f};
    for (int ic = 0; ic < 16; ++ic) {
        v8f base0 = {};
        v8f base1 = {};
#pragma unroll
        for (int c = 0; c < 16; c += 2) {
            const int l0 = c * 4 + hi * 2;
            const int l1 = (c + 1) * 4 + hi * 2;
            v2f bw0, bw1;
            bw0.x = sWz[l0 * IN_DIM + ic * 16 + m];
            bw0.y = sWz[(l0 + 1) * IN_DIM + ic * 16 + m];
            bw1.x = sWz[l1 * IN_DIM + ic * 16 + m];
            bw1.y = sWz[(l1 + 1) * IN_DIM + ic * 16 + m];
            base0 = wmma_f32(zf[c], bw0, base0);
            base1 = wmma_f32(zf[c + 1], bw1, base1);
        }
        const float bd = sBd[ic * 16 + m];
        const float* yc = y + (size_t)(row0 + hi * 8) * IN_DIM + ic * 16 + m;
#pragma unroll
        for (int r = 0; r < 8; ++r) {
            const float d = yc[(size_t)r * IN_DIM] - (base0[r] + base1[r]) - bd;
            sq[r] = fmaf(d, d, sq[r]);
        }
    }
    // reduce over the 16 N-lanes of each half -> broadcast row sums (C layout)
#pragma unroll
    for (int r = 0; r < 8; ++r) {
        float v = sq[r];
        v += __shfl_xor(v, 1, 32);
        v += __shfl_xor(v, 2, 32);
        v += __shfl_xor(v, 4, 32);
        v += __shfl_xor(v, 8, 32);
        sq[r] = v;
    }
    const float cc = sC[m];
#pragma unroll
    for (int r = 0; r < 8; ++r) {
        em[(size_t)(row0 + hi * 8 + r) * N_STATES + m] =
            (acc0[r] + acc1[r]) - 0.5f * sq[r] + cc;
    }
}

// ---------------------------------------------------------------------------
// Log-space recursion helpers.  Each chain lives in a 16-lane group (grp = 0
// or 16 within the wave); lane k of the group carries component k.
// ---------------------------------------------------------------------------
__device__ __forceinline__ void load_T(const float* __restrict__ trans, int k,
                                       float (&T)[16]) {
    for (int j = 0; j < 16; ++j) {
        float mj = trans[j * 16];
        for (int kk = 1; kk < 16; ++kk) mj = fmaxf(mj, trans[j * 16 + kk]);
        float sj = 0.0f;
        for (int kk = 0; kk < 16; ++kk) sj += __expf(trans[j * 16 + kk] - mj);
        T[j] = trans[j * 16 + k] - mj - __logf(sj);
    }
}

__device__ __forceinline__ float log_start_k(const float* __restrict__ start, int k) {
    float m0 = start[0];
#pragma unroll
    for (int j = 1; j < 16; ++j) m0 = fmaxf(m0, start[j]);
    float s0 = 0.0f;
#pragma unroll
    for (int j = 0; j < 16; ++j) s0 += __expf(start[j] - m0);
    return start[k] - m0 - __logf(s0);
}

// one step: p_k <- logsumexp_j(p_j + T[j,k]) + em_k
__device__ __forceinline__ float lse_step(float p, const float (&T)[16], int grp,
                                          float emk) {
    float x[16];
    float mx = -1e30f;
#pragma unroll
    for (int j = 0; j < 16; ++j) {
        x[j] = __shfl(p, grp + j, 32) + T[j];
        mx = fmaxf(mx, x[j]);
    }
    float s = 0.0f;
#pragma unroll
    for (int j = 0; j < 16; ++j) s += __expf(x[j] - mx);
    return mx + __logf(s) + emk;
}

// ---------------------------------------------------------------------------
// Phase 1: chunk products.  Chain id = ((b*NCHUNK + c)*16 + i).
//   c == 0 : real alpha chain from alpha_0 over t=1..CHUNK-1 (only i==0 stores)
//   c >= 1 : row i of P_c = product of L_t over the chunk, from log-identity
// Two independent chains per wave (16-lane groups).
// ---------------------------------------------------------------------------
__global__ void hmm_scan_chunks(const float* __restrict__ em,
                                const float* __restrict__ start_logits,
                                const float* __restrict__ trans_logits,
                                float* __restrict__ P) {
    const int lane  = threadIdx.x & 31;
    const int k     = lane & 15;
    const int grp   = lane & 16;
    const int gwave = (blockIdx.x * blockDim.x + threadIdx.x) >> 5;
    const int chain = (gwave << 1) | (grp >> 4);     // 0 .. B*NCHUNK*16-1
    const int i  = chain & 15;
    const int bc = chain >> 4;
    const int c  = bc & (NCHUNK - 1);
    const int b  = bc >> 5;                          // NCHUNK == 32

    float T[16];
    load_T(trans_logits, k, T);

    const float* eb = em + (size_t)b * SEQ * N_STATES;

    float p;
    int t0, t1;
    if (c == 0) {
        p = log_start_k(start_logits, k) + eb[k];    // alpha_0
        t0 = 1; t1 = CHUNK;
    } else {
        p = (k == i) ? 0.0f : -1e30f;                // log-identity row i
        t0 = c * CHUNK; t1 = t0 + CHUNK;
    }
    for (int t = t0; t < t1; ++t) {
        if (k == 0) __builtin_prefetch(eb + (size_t)(t + 16) * N_STATES, 0, 1);
        p = lse_step(p, T, grp, eb[(size_t)t * N_STATES + k]);
    }
    if (c == 0) {
        if (i == 0) P[((size_t)bc * 16 + 0) * 16 + k] = p;  // ending alpha of chunk 0
    } else {
        P[((size_t)bc * 16 + i) * 16 + k] = p;
    }
}

// ---------------------------------------------------------------------------
// Phase 2: serial boundary combine per batch (one wave per batch, lanes 0..15).
//   bnd_0 = ending alpha of chunk 0; bnd_c = bnd_{c-1} (x) P_c  (no extra em!)
// ---------------------------------------------------------------------------
__global__ void hmm_combine(const float* __restrict__ P, float* __restrict__ Bnd) {
    const int b = blockIdx.x;
    const int lane = threadIdx.x;   // lanes 16..31 mirror lanes 0..15
    const int k = lane & 15;
    const int grp = lane & 16;

    float bnd = P[(((size_t)b * NCHUNK + 0) * 16 + 0) * 16 + k];
    if (lane < 16) Bnd[((size_t)b * NCHUNK + 0) * 16 + k] = bnd;
    for (int c = 1; c < NCHUNK; ++c) {
        const float* Pc = P + ((size_t)b * NCHUNK + c) * 256;
        float x[16];
        float mx = -1e30f;
#pragma unroll
        for (int j = 0; j < 16; ++j) {
            x[j] = __shfl(bnd, grp + j, 32) + Pc[j * 16 + k];
            mx = fmaxf(mx, x[j]);
        }
        float s = 0.0f;
#pragma unroll
        for (int j = 0; j < 16; ++j) s += __expf(x[j] - mx);
        bnd = mx + __logf(s);
        if (lane < 16) Bnd[((size_t)b * NCHUNK + c) * 16 + k] = bnd;
    }
}

// ---------------------------------------------------------------------------
// Phase 3: per-chunk alpha replay + output.  Chain id = b*NCHUNK + c.
//   c == 0 : from alpha_0, write t = 0 .. CHUNK-1
//   c >= 1 : from Bnd[b,c-1], write t = c*CHUNK .. (c+1)*CHUNK-1
// Two chains per wave.
// ---------------------------------------------------------------------------
__global__ void hmm_alpha_chunks(const float* __restrict__ em,
                                 const float* __restrict__ start_logits,
                                 const float* __restrict__ trans_logits,
                                 const float* __restrict__ Bnd,
                                 float* __restrict__ alpha) {
    const int lane  = threadIdx.x & 31;
    const int k     = lane & 15;
    const int grp   = lane & 16;
    const int gwave = (blockIdx.x * blockDim.x + threadIdx.x) >> 5;
    const int chain = (gwave << 1) | (grp >> 4);     // 0 .. B*NCHUNK-1
    const int c = chain & (NCHUNK - 1);
    const int b = chain >> 5;

    float T[16];
    load_T(trans_logits, k, T);

    const float* eb = em + (size_t)b * SEQ * N_STATES;
    float* ab = alpha + (size_t)b * SEQ * N_STATES;

    float a;
    int t0;
    if (c == 0) {
        a = log_start_k(start_logits, k) + eb[k];
        ab[k] = a;
        t0 = 1;
    } else {
        a = Bnd[((size_t)b * NCHUNK + (c - 1)) * 16 + k];
        t0 = c * CHUNK;
    }
    const int t1 = (c + 1) * CHUNK;
    for (int t = t0; t < t1; ++t) {
        if (k == 0) __builtin_prefetch(eb + (size_t)(t + 16) * N_STATES, 0, 1);
        a = lse_step(a, T, grp, eb[(size_t)t * N_STATES + k]);
        ab[(size_t)t * N_STATES + k] = a;
    }
}

// ---------------------------------------------------------------------------
extern "C" void kernel_launch(void* const* d_in, const int* in_sizes, int n_in,
                              void* d_out, int out_size, void* d_ws, size_t ws_size,
                              hipStream_t stream) {
    (void)in_sizes; (void)n_in; (void)out_size; (void)ws_size;
    const float* y     = (const float*)d_in[0];
    const float* z     = (const float*)d_in[1];
    const float* start = (const float*)d_in[2];
    const float* trans = (const float*)d_in[3];
    const float* mu    = (const float*)d_in[4];
    const float* lv    = (const float*)d_in[5];
    const float* Wz    = (const float*)d_in[6];
    const float* Ws    = (const float*)d_in[7];
    const float* bdec  = (const float*)d_in[8];
    float* alpha = (float*)d_out;

    float* ws   = (float*)d_ws;
    float* em   = ws + WS_EM;
    float* Amat = ws + WS_AM;
    float* B2   = ws + WS_B2;
    float* Cc   = ws + WS_CC;
    float* P    = ws + WS_P;     // BATCH*NCHUNK*16*16
    float* Bnd  = ws + WS_BND;   // BATCH*NCHUNK*16

    hipLaunchKernelGGL(hmm_setup, dim3(1), dim3(1024), 0, stream,
                       mu, lv, Wz, Ws, bdec, Amat, B2, Cc);
    hipLaunchKernelGGL(hmm_emissions, dim3(EM_BLOCKS), dim3(256),
                       SMEM_FLOATS * sizeof(float), stream,
                       y, z, Wz, Ws, bdec, Amat, B2, Cc, em);
    // phase 1: B*NCHUNK*16 chains, 2 per wave, 16 chains per 256-thread block
    hipLaunchKernelGGL(hmm_scan_chunks, dim3(BATCH * NCHUNK * 16 / 16), dim3(256),
                       0, stream, em, start, trans, P);
    // phase 2: one wave per batch
    hipLaunchKernelGGL(hmm_combine, dim3(BATCH), dim3(32), 0, stream, P, Bnd);
    // phase 3: B*NCHUNK chains, 2 per wave, 16 chains per 256-thread block
    hipLaunchKernelGGL(hmm_alpha_chunks, dim3(BATCH * NCHUNK / 16), dim3(256),
                       0, stream, em, start, trans, Bnd, alpha);
}